// model_6811818132079
// MI455X (gfx1250) — compile-verified
//
#include <hip/hip_runtime.h>
#include <hip/hip_bf16.h>

// Problem constants (K=128, NR=1, NT=128, D=1, NLAYER=8)
#define NK 128
#define PLANE 16384            // 128*128
#define SQRT_PMAX 3.16227766016838f
#define SP 0.1f                // SIGMA/PMAX = 1/10

typedef __attribute__((ext_vector_type(2))) float v2f;
typedef __attribute__((ext_vector_type(8))) float v8f;

// ---------------------------------------------------------------------------
// Wave-level complex 128x128x128 GEMM helper using V_WMMA_F32_16X16X4_F32.
//   C = opA(A) * opB(B); re/im planes, row-major, ld=128.
//   TA/TB: operand is transpose of stored matrix.  CA/CB: conjugate operand.
//   TOUT: store C transposed (C[col][row]).  addDiag: add (dre,dim) on diag.
// Each of the 32 waves covers 2 of the 64 16x16 output tiles.
// Complex product via 4 real WMMA accumulators; conjugation applied as signs
// in the epilogue (F32 WMMA NEG only supports C-negate per ISA).
// ---------------------------------------------------------------------------
template <bool TA, bool CA, bool TB, bool CB, bool TOUT>
__device__ __forceinline__ void cgemm_tiles(
    const float* __restrict__ Ar, const float* __restrict__ Ai,
    const float* __restrict__ Br, const float* __restrict__ Bi,
    float* __restrict__ Cr, float* __restrict__ Ci,
    float dre, float dim, bool addDiag, int wave, int lane) {
  const int mr = lane & 15;          // A: M index / B: N index within tile
  const int kb = (lane >> 4) << 1;   // K sub-pair (0 or 2)
  for (int tile = wave; tile < 64; tile += 32) {
    const int tm = (tile >> 3) << 4;
    const int tn = (tile & 7) << 4;
    v8f p1 = {}, p2 = {}, p3 = {}, p4 = {};
    for (int k0 = 0; k0 < 128; k0 += 4) {
      const int ra = tm + mr;
      const int cn = tn + mr;
      const int c0 = k0 + kb;        // even -> 8B-aligned pair loads
      v2f ar, ai, br, bi;
      if (!TA) {                     // consecutive dwords -> b64 load
        ar = *(const v2f*)(Ar + ra * 128 + c0);
        ai = *(const v2f*)(Ai + ra * 128 + c0);
      } else {
        ar.x = Ar[c0 * 128 + ra]; ar.y = Ar[(c0 + 1) * 128 + ra];
        ai.x = Ai[c0 * 128 + ra]; ai.y = Ai[(c0 + 1) * 128 + ra];
      }
      if (TB) {                      // consecutive dwords -> b64 load
        br = *(const v2f*)(Br + cn * 128 + c0);
        bi = *(const v2f*)(Bi + cn * 128 + c0);
      } else {
        br.x = Br[c0 * 128 + cn]; br.y = Br[(c0 + 1) * 128 + cn];
        bi.x = Bi[c0 * 128 + cn]; bi.y = Bi[(c0 + 1) * 128 + cn];
      }
      p1 = __builtin_amdgcn_wmma_f32_16x16x4_f32(false, ar, false, br, (short)0, p1, false, false);
      p2 = __builtin_amdgcn_wmma_f32_16x16x4_f32(false, ai, false, bi, (short)0, p2, false, false);
      p3 = __builtin_amdgcn_wmma_f32_16x16x4_f32(false, ai, false, br, (short)0, p3, false, false);
      p4 = __builtin_amdgcn_wmma_f32_16x16x4_f32(false, ar, false, bi, (short)0, p4, false, false);
    }
    const float sA = CA ? -1.f : 1.f;
    const float sB = CB ? -1.f : 1.f;
    #pragma unroll
    for (int j = 0; j < 8; ++j) {
      const int row = tm + ((lane >> 4) << 3) + j;
      const int col = tn + (lane & 15);
      float cre = p1[j] - sA * sB * p2[j];
      float cim = sA * p3[j] + sB * p4[j];
      if (addDiag && row == col) { cre += dre; cim += dim; }
      const int o = TOUT ? (col * 128 + row) : (row * 128 + col);
      Cr[o] = cre;
      Ci[o] = cim;
    }
  }
}

// ---------------------------------------------------------------------------
// Fused per-layer pipeline: one block, 1024 threads (32 waves, one WGP).
//  phase 0: normalize incoming V (skip on first layer)
//  phase 1: S = V^T conj(V)          -> M1   (WMMA)
//  phase 2: T = H S                  -> M2   (WMMA)
//  phase 3: per-k scalar chain, G = diag(c) H -> M3, uw -> sc/LDS, trUWU
//  phase 4: B = 0.1 trUWU I + H^H G  -> M1   (WMMA)
//  phase 5: Gauss-Jordan inversion of B in LDS (HPD, no pivoting) -> M2
//  phase 6: YT[k][t] = (Binv H^H)^T  -> M3   (WMMA, transposed store)
//  phase 7 (final layer): V = YT*uw, normalize, write d_out
// ---------------------------------------------------------------------------
__global__ __launch_bounds__(1024) void fused_layer(
    const float* __restrict__ Hr, const float* __restrict__ Hi,
    float* __restrict__ Vr, float* __restrict__ Vi,
    float* __restrict__ M1r, float* __restrict__ M1i,
    float* __restrict__ M2r, float* __restrict__ M2i,
    float* __restrict__ M3r, float* __restrict__ M3i,
    const float* __restrict__ up1, const float* __restrict__ up3,
    const float* __restrict__ up4, const float* __restrict__ wp1,
    const float* __restrict__ wp3,
    float* __restrict__ sc, int normIn, int isFinal,
    float* __restrict__ outAll) {
  __shared__ float Bs_r[PLANE];      // 64 KB  (B / Binv, LDS-resident)
  __shared__ float Bs_i[PLANE];      // 64 KB
  __shared__ float redbuf[1024];
  __shared__ float cre_s[NK], cim_s[NK];
  __shared__ float uwr_s[NK], uwi_s[NK];
  __shared__ float prw[NK], piw[NK], fcr[NK], fci[NK];
  const int tid = threadIdx.x;
  const int wave = tid >> 5;
  const int lane = tid & 31;

  auto blockReduce = [&](float v) -> float {
    redbuf[tid] = v; __syncthreads();
    for (int s = 512; s > 0; s >>= 1) {
      if (tid < s) redbuf[tid] += redbuf[tid + s];
      __syncthreads();
    }
    const float r = redbuf[0];
    __syncthreads();
    return r;
  };

  // ---- phase 0: normalize incoming V
  if (normIn) {
    float acc = 0.f;
    for (int i = tid; i < PLANE; i += 1024) acc += Vr[i] * Vr[i] + Vi[i] * Vi[i];
    const float s = SQRT_PMAX * rsqrtf(blockReduce(acc));
    for (int i = tid; i < PLANE; i += 1024) { Vr[i] *= s; Vi[i] *= s; }
    __syncthreads();
  }

  // ---- phase 1: S[n][m] = sum_k V[k][n] conj(V[k][m])
  cgemm_tiles<true, false, false, true, false>(Vr, Vi, Vr, Vi, M1r, M1i,
                                               0.f, 0.f, false, wave, lane);
  __syncthreads();
  // ---- phase 2: T = H * S
  cgemm_tiles<false, false, false, false, false>(Hr, Hi, M1r, M1i, M2r, M2i,
                                                 0.f, 0.f, false, wave, lane);
  __syncthreads();

  // ---- phase 3: scalar chain
  float acc = 0.f;
  for (int i = tid; i < PLANE; i += 1024) acc += Vr[i] * Vr[i] + Vi[i] * Vi[i];
  const float trvv = blockReduce(acc);
  for (int kk = 0; kk < 4; ++kk) {
    const int k = wave + 32 * kk;
    float da_r = 0.f, da_i = 0.f, hv_r = 0.f, hv_i = 0.f;
    for (int m = lane; m < 128; m += 32) {
      const float tr = M2r[k * 128 + m], ti = M2i[k * 128 + m];
      const float hr = Hr[k * 128 + m], hi = Hi[k * 128 + m];
      da_r += tr * hr + ti * hi;                 // T * conj(H)
      da_i += ti * hr - tr * hi;
      const float vr = Vr[k * 128 + m], vi = Vi[k * 128 + m];
      hv_r += hr * vr - hi * vi;                 // H * V
      hv_i += hr * vi + hi * vr;
    }
    for (int off = 16; off > 0; off >>= 1) {
      da_r += __shfl_down(da_r, off, 32);
      da_i += __shfl_down(da_i, off, 32);
      hv_r += __shfl_down(hv_r, off, 32);
      hv_i += __shfl_down(hv_i, off, 32);
    }
    if (lane == 0) {
      const float A_re = SP * trvv + da_r, A_im = da_i;
      float d = A_re * A_re + A_im * A_im;
      const float iAr = A_re / d, iAi = -A_im / d;
      const float u1r = up1[k], u1i = up1[128 + k];
      const float air = u1r * iAr - u1i * iAi + 0.01f * up3[k];
      const float aii = u1r * iAi + u1i * iAr + 0.01f * up3[128 + k];
      const float ur = air * hv_r - aii * hv_i + up4[k];
      const float ui = air * hv_i + aii * hv_r + up4[128 + k];
      const float er = 1.f - (ur * hv_r + ui * hv_i);   // 1 - conj(u)*hv
      const float ei = -(ur * hv_i - ui * hv_r);
      d = er * er + ei * ei;
      const float ier = er / d, iei = -ei / d;
      const float w1r = wp1[k], w1i = wp1[128 + k];
      const float wr = 0.1f * (w1r * ier - w1i * iei) + 0.01f * wp3[k];
      const float wi = 0.1f * (w1r * iei + w1i * ier) + 0.01f * wp3[128 + k];
      const float uwr = ur * wr - ui * wi;       // uw = u*W
      const float uwi = ur * wi + ui * wr;
      uwr_s[k] = uwr; uwi_s[k] = uwi;
      sc[k] = uwr; sc[128 + k] = uwi;            // for streaming V-update kernel
      const float u2 = ur * ur + ui * ui;        // c = |u|^2 W
      cre_s[k] = u2 * wr;
      cim_s[k] = u2 * wi;
    }
  }
  __syncthreads();
  const float trU_r = blockReduce(tid < 128 ? cre_s[tid] : 0.f);
  const float trU_i = blockReduce(tid < 128 ? cim_s[tid] : 0.f);
  // G[k][s] = c_k * H[k][s]  -> M3
  for (int i = tid; i < PLANE; i += 1024) {
    const int k = i >> 7;
    const float cr = cre_s[k], ci = cim_s[k];
    const float hr = Hr[i], hi = Hi[i];
    M3r[i] = cr * hr - ci * hi;
    M3i[i] = cr * hi + ci * hr;
  }
  __syncthreads();

  // ---- phase 4: B = 0.1*trUWU*I + H^H G  -> M1
  cgemm_tiles<true, true, false, false, false>(Hr, Hi, M3r, M3i, M1r, M1i,
                                               SP * trU_r, SP * trU_i, true,
                                               wave, lane);
  __syncthreads();

  // ---- phase 5: Gauss-Jordan inversion in LDS
  for (int i = tid; i < PLANE; i += 1024) { Bs_r[i] = M1r[i]; Bs_i[i] = M1i[i]; }
  __syncthreads();
  for (int p = 0; p < 128; ++p) {
    if (tid < 128) { fcr[tid] = Bs_r[tid * 128 + p]; fci[tid] = Bs_i[tid * 128 + p]; }
    __syncthreads();
    const float ar = fcr[p], ai = fci[p];
    const float d = ar * ar + ai * ai;
    const float qr = ar / d, qi = -ai / d;       // 1/a_pp
    if (tid < 128) {
      float xr, xi;
      if (tid == p) { xr = qr; xi = qi; }
      else {
        const float br = Bs_r[p * 128 + tid], bi = Bs_i[p * 128 + tid];
        xr = qr * br - qi * bi;
        xi = qr * bi + qi * br;
      }
      prw[tid] = xr; piw[tid] = xi;
      Bs_r[p * 128 + tid] = xr; Bs_i[p * 128 + tid] = xi;
    }
    __syncthreads();
    for (int idx = tid; idx < PLANE; idx += 1024) {
      const int i = idx >> 7, j = idx & 127;
      if (i == p) continue;
      const float cr = fcr[i], ci = fci[i];
      if (j == p) {
        Bs_r[idx] = -(cr * prw[p] - ci * piw[p]);
        Bs_i[idx] = -(cr * piw[p] + ci * prw[p]);
      } else {
        Bs_r[idx] -= cr * prw[j] - ci * piw[j];
        Bs_i[idx] -= cr * piw[j] + ci * prw[j];
      }
    }
    __syncthreads();
  }
  for (int i = tid; i < PLANE; i += 1024) { M2r[i] = Bs_r[i]; M2i[i] = Bs_i[i]; }
  __syncthreads();

  // ---- phase 6: YT[k][t] = sum_s Binv[t][s] conj(H[k][s])  (transposed store)
  cgemm_tiles<false, false, true, true, true>(M2r, M2i, Hr, Hi, M3r, M3i,
                                              0.f, 0.f, false, wave, lane);
  __syncthreads();

  // ---- phase 7: final layer tail: V = YT*uw, normalize, emit output
  if (isFinal) {
    for (int i = tid; i < PLANE; i += 1024) {
      const int k = i >> 7;
      const float yr = M3r[i], yi = M3i[i];
      const float ur = uwr_s[k], ui = uwi_s[k];
      Vr[i] = yr * ur - yi * ui;
      Vi[i] = yr * ui + yi * ur;
    }
    __syncthreads();
    float a2 = 0.f;
    for (int i = tid; i < PLANE; i += 1024) a2 += Vr[i] * Vr[i] + Vi[i] * Vi[i];
    const float s = SQRT_PMAX * rsqrtf(blockReduce(a2));
    for (int i = tid; i < PLANE; i += 1024) {
      outAll[i] = Vr[i] * s;
      outAll[PLANE + i] = Vi[i] * s;
    }
  }
}

// ---------------------------------------------------------------------------
// Mid-layer V update: the bandwidth kernel (streams mid_v_p2/p4, 33 MB/layer).
//   Vn[k][t] = (sum_s p2[k][t][s]*YT[k][s] + 0.01*sum_s p4[k][t][s]*conj(H[k][s]))
//              * uw_k + p1[k][t]          (Vn left unnormalized; next fused
//                                          kernel normalizes first)
// One wave per output element; lanes stride over s -> fully coalesced.
// ---------------------------------------------------------------------------
__global__ __launch_bounds__(256) void vupdate_mid(
    const float* __restrict__ Hr, const float* __restrict__ Hi,
    const float* __restrict__ Ytr, const float* __restrict__ Yti,
    const float* __restrict__ p2, const float* __restrict__ p4,
    const float* __restrict__ p1, const float* __restrict__ sc,
    float* __restrict__ Vr, float* __restrict__ Vi) {
  const int wave = blockIdx.x * 8 + (threadIdx.x >> 5);
  const int lane = threadIdx.x & 31;
  const int k = wave >> 7, t = wave & 127;
  const size_t rowOff = ((size_t)k * 128 + t) * 128;
  const float* p2r = p2 + rowOff;
  const float* p2i = p2 + 2097152 + rowOff;   // + K*NT*NT
  const float* p4r = p4 + rowOff;
  const float* p4i = p4 + 2097152 + rowOff;
  float ar = 0.f, ai = 0.f;
  for (int s = lane; s < 128; s += 32) {
    const float yr = Ytr[k * 128 + s], yi = Yti[k * 128 + s];
    const float a2r = p2r[s], a2i = p2i[s];
    ar += a2r * yr - a2i * yi;
    ai += a2r * yi + a2i * yr;
    const float hr = Hr[k * 128 + s], hi = -Hi[k * 128 + s];  // conj(H)
    const float a4r = p4r[s], a4i = p4i[s];
    ar += 0.01f * (a4r * hr - a4i * hi);
    ai += 0.01f * (a4r * hi + a4i * hr);
  }
  for (int off = 16; off > 0; off >>= 1) {
    ar += __shfl_down(ar, off, 32);
    ai += __shfl_down(ai, off, 32);
  }
  if (lane == 0) {
    const float uwr = sc[k], uwi = sc[128 + k];
    Vr[k * 128 + t] = ar * uwr - ai * uwi + p1[k * 128 + t];
    Vi[k * 128 + t] = ar * uwi + ai * uwr + p1[PLANE + k * 128 + t];
  }
}

// ---------------------------------------------------------------------------
// Host orchestration: 17 kernels + 1 d2d copy total.
// Workspace layout (floats):
//   [0] Vr [16384] Vi | [32768] M1r [49152] M1i | [65536] M2r [81920] M2i
//   [98304] M3r [114688] M3i | [131072] sc (uw re/im, 256 floats)
// ---------------------------------------------------------------------------
extern "C" void kernel_launch(void* const* d_in, const int* in_sizes, int n_in,
                              void* d_out, int out_size, void* d_ws, size_t ws_size,
                              hipStream_t stream) {
  (void)in_sizes; (void)n_in; (void)out_size; (void)ws_size;
  float* ws = (float*)d_ws;
  float* Vr = ws;            float* Vi = ws + PLANE;
  float* M1r = ws + 32768;   float* M1i = ws + 49152;
  float* M2r = ws + 65536;   float* M2i = ws + 81920;
  float* M3r = ws + 98304;   float* M3i = ws + 114688;
  float* sc  = ws + 131072;

  const float* Hr = (const float*)d_in[1];
  const float* Hi = Hr + PLANE;

  // V <- input V (re plane then im plane, contiguous)
  hipMemcpyAsync(Vr, d_in[0], 2 * PLANE * sizeof(float),
                 hipMemcpyDeviceToDevice, stream);

  for (int l = 0; l < 9; ++l) {
    const bool fin = (l == 8);
    const float* up1 = fin ? (const float*)d_in[10] : (const float*)d_in[2] + l * 256;
    const float* up3 = fin ? (const float*)d_in[11] : (const float*)d_in[3] + l * 256;
    const float* up4 = fin ? (const float*)d_in[12] : (const float*)d_in[4] + l * 256;
    const float* wp1 = fin ? (const float*)d_in[13] : (const float*)d_in[5] + l * 256;
    const float* wp3 = fin ? (const float*)d_in[14] : (const float*)d_in[6] + l * 256;

    fused_layer<<<1, 1024, 0, stream>>>(
        Hr, Hi, Vr, Vi, M1r, M1i, M2r, M2i, M3r, M3i,
        up1, up3, up4, wp1, wp3, sc,
        /*normIn=*/(l > 0) ? 1 : 0, /*isFinal=*/fin ? 1 : 0,
        fin ? (float*)d_out : nullptr);

    if (!fin) {
      const float* p2 = (const float*)d_in[8] + (size_t)l * 4194304;
      const float* p4 = (const float*)d_in[9] + (size_t)l * 4194304;
      const float* p1 = (const float*)d_in[7] + (size_t)l * 32768;
      vupdate_mid<<<2048, 256, 0, stream>>>(Hr, Hi, M3r, M3i, p2, p4, p1,
                                            sc, Vr, Vi);
    }
  }
}